// GAT_15728170238521
// MI455X (gfx1250) — compile-verified
//
#include <hip/hip_runtime.h>
#include <math.h>

#define IN_C 128
#define HID 32
#define HEADS 4
#define OUT_C 64
#define NEG_SLOPE 0.2f

typedef __attribute__((ext_vector_type(2))) float v2f;
typedef __attribute__((ext_vector_type(4))) float v4f;
typedef __attribute__((ext_vector_type(8))) float v8f;

// ------------------------------------------------------------------
// GEMM C[M,NC] = A[M,K] * B[K,NC], fp32, via V_WMMA_F32_16X16X4_F32.
// One wave (32 threads) computes one 16x16 tile of C.
// A 16x4 layout : lanes 0-15 hold K={0,1}, lanes 16-31 hold K={2,3}, M=lane%16
// B 4x16 layout : same half-wave K split, N=lane%16
// C 16x16 layout: VGPR i -> M = i + 8*(lane>=16), N = lane%16
// ------------------------------------------------------------------
template <int K, int NC>
__global__ void gemm_wmma(const float* __restrict__ A, const float* __restrict__ B,
                          float* __restrict__ C, int M) {
  const int lane  = threadIdx.x;       // 0..31
  const int m16   = lane & 15;
  const int khalf = lane >> 4;         // 0 or 1
  int row = blockIdx.x * 16 + m16;
  if (row >= M) row = M - 1;           // clamp loads; EXEC stays all-ones for WMMA
  const int col = (int)blockIdx.y * 16 + m16;
  const float* arow = A + (size_t)row * K;

  v8f acc = {};
#pragma unroll 4
  for (int k0 = 0; k0 < K; k0 += 4) {
    const int ka = k0 + khalf * 2;
    v2f av, bv;
    av.x = arow[ka];
    av.y = arow[ka + 1];
    bv.x = B[ka * NC + col];
    bv.y = B[(ka + 1) * NC + col];
    acc = __builtin_amdgcn_wmma_f32_16x16x4_f32(false, av, false, bv, (short)0,
                                                acc, false, false);
  }

  const int rbase = (int)blockIdx.x * 16 + khalf * 8;
  float* cbase = C + (size_t)rbase * NC + col;
  if ((int)blockIdx.x * 16 + 16 <= M) {
    // full tile: unguarded stores, immediate offsets from one base
#pragma unroll
    for (int i = 0; i < 8; ++i) cbase[(size_t)i * NC] = acc[i];
  } else {
#pragma unroll
    for (int i = 0; i < 8; ++i)
      if (rbase + i < M) cbase[(size_t)i * NC] = acc[i];
  }
}

// ------------------------------------------------------------------
// Per-node attention logits: als[n,h] = dot(h[n,h,:], a_src[h,:]) etc.
// ------------------------------------------------------------------
template <int H, int CHPH>
__global__ void node_logits(const float* __restrict__ h, const float* __restrict__ asrc,
                            const float* __restrict__ adst,
                            float* __restrict__ als, float* __restrict__ ald, int NH) {
  int tid = blockIdx.x * blockDim.x + threadIdx.x;
  if (tid >= NH) return;
  int node = tid / H;
  int head = tid % H;
  const float* hp = h + (size_t)node * (H * CHPH) + head * CHPH;
  const float* as = asrc + head * CHPH;
  const float* ad = adst + head * CHPH;
  float s = 0.f, d = 0.f;
#pragma unroll
  for (int c = 0; c < CHPH; ++c) {
    float v = hp[c];
    s = fmaf(v, as[c], s);
    d = fmaf(v, ad[c], d);
  }
  als[tid] = s;
  ald[tid] = d;
}

// Monotonic order-preserving float<->uint key (for atomic max on floats).
__device__ __forceinline__ unsigned fkey(float f) {
  unsigned u = __float_as_uint(f);
  return (u & 0x80000000u) ? ~u : (u | 0x80000000u);
}
__device__ __forceinline__ float funkey(unsigned k) {
  return __uint_as_float((k & 0x80000000u) ? (k & 0x7fffffffu) : ~k);
}

__device__ __forceinline__ float leaky(float v) {
  return v > 0.f ? v : NEG_SLOPE * v;
}

template <int H>
__global__ void edge_max_k(const int* __restrict__ src, const int* __restrict__ dst,
                           const float* __restrict__ als, const float* __restrict__ ald,
                           unsigned* __restrict__ mkey, int EH) {
  int tid = blockIdx.x * blockDim.x + threadIdx.x;
  if (tid >= EH) return;
  int e = tid / H;
  int head = tid % H;
  int s = src[e], d = dst[e];
  float lr = leaky(als[s * H + head] + ald[d * H + head]);
  atomicMax(&mkey[d * H + head], fkey(lr));
}

template <int H>
__global__ void edge_expsum_k(const int* __restrict__ src, const int* __restrict__ dst,
                              const float* __restrict__ als, const float* __restrict__ ald,
                              const unsigned* __restrict__ mkey, float* __restrict__ ssum,
                              int EH) {
  int tid = blockIdx.x * blockDim.x + threadIdx.x;
  if (tid >= EH) return;
  int e = tid / H;
  int head = tid % H;
  int s = src[e], d = dst[e];
  float lr = leaky(als[s * H + head] + ald[d * H + head]);
  float ex = __expf(lr - funkey(mkey[d * H + head]));
  atomicAdd(&ssum[d * H + head], ex);
}

// One 32-lane wave per edge; lane owns CH/32 contiguous channels (single head).
template <int H, int CH>
__global__ void edge_aggregate(const int* __restrict__ src, const int* __restrict__ dst,
                               const float* __restrict__ h,
                               const float* __restrict__ als, const float* __restrict__ ald,
                               const unsigned* __restrict__ mkey, const float* __restrict__ ssum,
                               float* __restrict__ out, int E) {
  int tid = blockIdx.x * blockDim.x + threadIdx.x;
  int e = tid >> 5;
  int lane = tid & 31;
  if (e >= E) return;
  int s = src[e], d = dst[e];
  constexpr int PER = CH / 32;    // channels per lane (4 or 2)
  constexpr int CHPH = CH / H;    // channels per head
  const int c0 = lane * PER;
  const int head = c0 / CHPH;     // constant per lane
  float lr = leaky(als[s * H + head] + ald[d * H + head]);
  float w = __expf(lr - funkey(mkey[d * H + head])) /
            (ssum[d * H + head] + 1e-16f);
  const float* hrow = h + (size_t)s * CH + c0;
  float* orow = out + (size_t)d * CH + c0;
  if (PER == 4) {
    v4f hv = *(const v4f*)hrow;   // global_load_b128 gather
    atomicAdd(&orow[0], hv.x * w);
    atomicAdd(&orow[1], hv.y * w);
    atomicAdd(&orow[2], hv.z * w);
    atomicAdd(&orow[3], hv.w * w);
  } else {
    v2f hv = *(const v2f*)hrow;   // global_load_b64 gather
    atomicAdd(&orow[0], hv.x * w);
    atomicAdd(&orow[1], hv.y * w);
  }
}

__global__ void fill_u32(unsigned* __restrict__ p, unsigned v, int n) {
  int i = blockIdx.x * blockDim.x + threadIdx.x;
  if (i < n) p[i] = v;
}

template <int CH, bool ELU>
__global__ void bias_act(float* __restrict__ x, const float* __restrict__ b, int n) {
  int i = blockIdx.x * blockDim.x + threadIdx.x;
  if (i >= n) return;
  int c = i % CH;
  float v = x[i] + b[c];
  if (ELU) v = v > 0.f ? v : (__expf(v) - 1.0f);
  x[i] = v;
}

extern "C" void kernel_launch(void* const* d_in, const int* in_sizes, int n_in,
                              void* d_out, int out_size, void* d_ws, size_t ws_size,
                              hipStream_t stream) {
  const float* x   = (const float*)d_in[0];
  const int*   ei  = (const int*)d_in[1];
  const float* W1  = (const float*)d_in[2];
  const float* as1 = (const float*)d_in[3];
  const float* ad1 = (const float*)d_in[4];
  const float* b1  = (const float*)d_in[5];
  const float* W2  = (const float*)d_in[6];
  const float* as2 = (const float*)d_in[7];
  const float* ad2 = (const float*)d_in[8];
  const float* b2  = (const float*)d_in[9];
  float* out = (float*)d_out;

  const int N = in_sizes[0] / IN_C;
  const int E = in_sizes[1] / 2;
  const int* src = ei;
  const int* dst = ei + E;

  // Workspace layout (floats). Layer 2 reuses layer-1 buffers.
  float* ws = (float*)d_ws;
  float*    h1   = ws;                                    // N*128 (layer2: h2 uses first N*64)
  float*    agg1 = h1 + (size_t)N * 128;                  // N*128 (becomes act1 in place)
  float*    als  = agg1 + (size_t)N * 128;                // N*HEADS
  float*    ald  = als + (size_t)N * HEADS;               // N*HEADS
  unsigned* mkey = (unsigned*)(ald + (size_t)N * HEADS);  // N*HEADS
  float*    ssum = (float*)(mkey + (size_t)N * HEADS);    // N*HEADS

  auto cdiv = [](long long a, long long b) -> unsigned { return (unsigned)((a + b - 1) / b); };
  const unsigned KEY_NEG_INF = 0x007FFFFFu;  // fkey(-inf): minimum key

  // ================= Layer 1 (4 heads x 32 ch) =================
  gemm_wmma<IN_C, HEADS * HID>
      <<<dim3(cdiv(N, 16), (HEADS * HID) / 16), 32, 0, stream>>>(x, W1, h1, N);
  node_logits<HEADS, HID>
      <<<cdiv((long long)N * HEADS, 256), 256, 0, stream>>>(h1, as1, ad1, als, ald, N * HEADS);
  fill_u32<<<cdiv((long long)N * HEADS, 256), 256, 0, stream>>>(mkey, KEY_NEG_INF, N * HEADS);
  fill_u32<<<cdiv((long long)N * HEADS, 256), 256, 0, stream>>>((unsigned*)ssum, 0u, N * HEADS);
  fill_u32<<<cdiv((long long)N * 128, 256), 256, 0, stream>>>((unsigned*)agg1, 0u, N * 128);
  edge_max_k<HEADS>
      <<<cdiv((long long)E * HEADS, 256), 256, 0, stream>>>(src, dst, als, ald, mkey, E * HEADS);
  edge_expsum_k<HEADS>
      <<<cdiv((long long)E * HEADS, 256), 256, 0, stream>>>(src, dst, als, ald, mkey, ssum,
                                                            E * HEADS);
  edge_aggregate<HEADS, 128>
      <<<cdiv((long long)E * 32, 256), 256, 0, stream>>>(src, dst, h1, als, ald, mkey, ssum,
                                                         agg1, E);
  bias_act<128, true><<<cdiv((long long)N * 128, 256), 256, 0, stream>>>(agg1, b1, N * 128);

  // ================= Layer 2 (1 head x 64 ch) =================
  float* h2 = h1;  // reuse (N*64)
  gemm_wmma<HEADS * HID, OUT_C>
      <<<dim3(cdiv(N, 16), OUT_C / 16), 32, 0, stream>>>(agg1, W2, h2, N);
  node_logits<1, OUT_C><<<cdiv(N, 256), 256, 0, stream>>>(h2, as2, ad2, als, ald, N);
  fill_u32<<<cdiv(N, 256), 256, 0, stream>>>(mkey, KEY_NEG_INF, N);
  fill_u32<<<cdiv(N, 256), 256, 0, stream>>>((unsigned*)ssum, 0u, N);
  fill_u32<<<cdiv((long long)N * OUT_C, 256), 256, 0, stream>>>((unsigned*)out, 0u, N * OUT_C);
  edge_max_k<1><<<cdiv(E, 256), 256, 0, stream>>>(src, dst, als, ald, mkey, E);
  edge_expsum_k<1><<<cdiv(E, 256), 256, 0, stream>>>(src, dst, als, ald, mkey, ssum, E);
  edge_aggregate<1, OUT_C>
      <<<cdiv((long long)E * 32, 256), 256, 0, stream>>>(src, dst, h2, als, ald, mkey, ssum,
                                                         out, E);
  bias_act<OUT_C, false><<<cdiv((long long)N * OUT_C, 256), 256, 0, stream>>>(out, b2, N * OUT_C);
}